// EulerAttentionHead_75892072121093
// MI455X (gfx1250) — compile-verified
//
#include <hip/hip_runtime.h>
#include <math.h>

// ---------------- problem constants ----------------
#define B_DIM   32
#define S_DIM   4096
#define D_DIM   2048
#define DH      128

#define PHI_F        1.6180339887498949f
#define Q2I          651.8986469044033f      // LUT_SIZE / (2*pi) = 4096/(2pi)
#define I2Q          0.0015339807878856412f  // (2*pi) / LUT_SIZE
#define INV_SCALE    0.0625f                 // 1/sqrt(2*128) = 1/16

typedef float v2f __attribute__((ext_vector_type(2)));
typedef float v8f __attribute__((ext_vector_type(8)));

// sin_tab[floor(theta*Q2I) mod 4096] == sinf(floor(theta*Q2I)*I2Q) by periodicity,
// so the LUT reduces to quantizing theta to the grid.
__device__ __forceinline__ float quantize_theta(float theta) {
    return floorf(theta * Q2I) * I2Q;
}

// ---------------------------------------------------------------------------
// Kernel A: query[b, 0:256] = [cos(q(theta_q)), sin(q(theta_q))]
// grid = 32 blocks (one per b), 128 threads (one per head dim)
// ---------------------------------------------------------------------------
__global__ void ea_query_kernel(const float* __restrict__ x,
                                const float* __restrict__ t,
                                const float* __restrict__ wq,
                                const float* __restrict__ bq,
                                float* __restrict__ qbuf) {
    const int b = blockIdx.x;
    const int d = threadIdx.x;
    float theta = x[(size_t)b * D_DIM + d] / (1.0f + fabsf(wq[d])) + bq[d] + t[b] * PHI_F;
    float qt = quantize_theta(theta);
    float sv, cv;
    __sincosf(qt, &sv, &cv);
    qbuf[b * 256 + d]       = cv;   // cos half of query
    qbuf[b * 256 + 128 + d] = sv;   // sin half of query
}

// ---------------------------------------------------------------------------
// Kernel B: raw scores[b,s] = (sum_d cq*ck + sq*sk) / 16
// One wave per group of 8 s-positions; lanes stride over the 128 head dims
// (4 coalesced 128B loads per s-row). grid = 32*64 blocks of 256 threads.
// ---------------------------------------------------------------------------
__global__ void ea_scores_kernel(const float* __restrict__ cached,
                                 const float* __restrict__ wk,
                                 const float* __restrict__ bk,
                                 const float* __restrict__ qbuf,
                                 float* __restrict__ scores) {
    const int b     = blockIdx.x >> 6;          // 64 blocks per batch row
    const int sblk  = blockIdx.x & 63;
    const int wave  = threadIdx.x >> 5;
    const int lane  = threadIdx.x & 31;
    const int sbase = sblk * 64 + wave * 8;

    // per-lane key params + query values for its 4 head dims (hoisted)
    float invwk[4], bkk[4], cq[4], sq[4];
#pragma unroll
    for (int k = 0; k < 4; ++k) {
        const int d = lane + 32 * k;
        invwk[k] = 1.0f / (1.0f + fabsf(wk[d]));
        bkk[k]   = bk[d];
        cq[k]    = qbuf[b * 256 + d];
        sq[k]    = qbuf[b * 256 + 128 + d];
    }

    const float* base = cached + (size_t)b * S_DIM * D_DIM;
    for (int j = 0; j < 8; ++j) {
        const int s = sbase + j;
        const float* row = base + (size_t)s * D_DIM;
        float acc = 0.0f;
#pragma unroll
        for (int k = 0; k < 4; ++k) {
            const float v  = row[lane + 32 * k];            // coalesced 128B per load
            const float qt = quantize_theta(fmaf(v, invwk[k], bkk[k]));
            float sk, ck;
            __sincosf(qt, &sk, &ck);
            acc = fmaf(ck, cq[k], acc);
            acc = fmaf(sk, sq[k], acc);
        }
        // wave32 butterfly reduce
#pragma unroll
        for (int off = 16; off > 0; off >>= 1)
            acc += __shfl_xor(acc, off, 32);
        if (lane == 0) scores[b * S_DIM + s] = acc * INV_SCALE;
    }
}

// ---------------------------------------------------------------------------
// Kernel C: in-place softmax over s (4096) per batch row. 32 blocks x 256.
// ---------------------------------------------------------------------------
__global__ void ea_softmax_kernel(float* __restrict__ scores) {
    __shared__ float red[256];
    const int b   = blockIdx.x;
    const int tid = threadIdx.x;
    float* row = scores + b * S_DIM;

    float m = -INFINITY;
    for (int i = tid; i < S_DIM; i += 256) m = fmaxf(m, row[i]);
    red[tid] = m; __syncthreads();
    for (int off = 128; off > 0; off >>= 1) {
        if (tid < off) red[tid] = fmaxf(red[tid], red[tid + off]);
        __syncthreads();
    }
    m = red[0]; __syncthreads();

    float sum = 0.0f;
    for (int i = tid; i < S_DIM; i += 256) sum += __expf(row[i] - m);
    red[tid] = sum; __syncthreads();
    for (int off = 128; off > 0; off >>= 1) {
        if (tid < off) red[tid] += red[tid + off];
        __syncthreads();
    }
    const float inv = 1.0f / red[0];

    for (int i = tid; i < S_DIM; i += 256) row[i] = __expf(row[i] - m) * inv;
}

// ---------------------------------------------------------------------------
// Kernel D: output[b,d] = sum_s w[b,s] * cached[b,s,d]  via V_WMMA_F32_16X16X4_F32.
//
// Per wave: one 16-wide d-tile, K-loop over s in chunks of 4.
//   A (16x4, MxK): weights w[s0..s0+3] replicated across all M rows.
//     Layout (32-bit A 16x4): lanes 0-15 hold {K0,K1}, lanes 16-31 hold {K2,K3}
//     -> lane reads wlds[s0+2*half .. +1] as one ds_load_b64 (all 16 rows equal).
//   B (4x16, KxN): cached tile; lane (half,col) loads V[s0+2*half+{0,1}, d0+col]
//     -> 64B-contiguous across lanes per row (coalesced).
//   D: all 16 rows identical; row M=0 lives in acc[0] of lanes 0-15.
// Block = 8 waves (128 d) sharing a 16KB LDS copy of the weight row.
// grid = 32 b * 16 d-groups = 512 blocks; each block streams a disjoint 2MB
// slice of cached_states -> exactly one pass over the 1.07GB tensor.
// ---------------------------------------------------------------------------
__global__ void ea_output_wmma_kernel(const float* __restrict__ cached,
                                      const float* __restrict__ weights,
                                      float* __restrict__ out) {
    __shared__ float wlds[S_DIM];               // 16 KB
    const int b    = blockIdx.x >> 4;
    const int dgrp = blockIdx.x & 15;
    const int tid  = threadIdx.x;
    const int wave = tid >> 5;
    const int lane = tid & 31;
    const int half = lane >> 4;                 // 0: K rows {0,1}; 1: K rows {2,3}
    const int col  = lane & 15;                 // N index

    for (int i = tid; i < S_DIM; i += 256)
        wlds[i] = weights[b * S_DIM + i];
    __syncthreads();

    const int d0 = dgrp * 128 + wave * 16;
    const float* base = cached + (size_t)b * S_DIM * D_DIM + (size_t)(d0 + col);
    const int srow = 2 * half;

    v8f acc = {};
#pragma unroll 4
    for (int s0 = 0; s0 < S_DIM; s0 += 4) {
        const float* p = base + (size_t)(s0 + srow) * D_DIM;
        v2f a, bm;
        a.x  = wlds[s0 + srow];                 // 8B-aligned pair -> ds_load_b64
        a.y  = wlds[s0 + srow + 1];
        bm.x = p[0];                            // V[s0+srow  , d0+col]
        bm.y = p[D_DIM];                        // V[s0+srow+1, d0+col]
        acc = __builtin_amdgcn_wmma_f32_16x16x4_f32(
            /*neg_a=*/false, a, /*neg_b=*/false, bm,
            /*c_mod=*/(short)0, acc, /*reuse_a=*/false, /*reuse_b=*/false);
    }

    if (lane < 16)                              // row M=0, N=lane
        out[(size_t)b * D_DIM + d0 + lane] = acc[0];
}

// ---------------------------------------------------------------------------
extern "C" void kernel_launch(void* const* d_in, const int* in_sizes, int n_in,
                              void* d_out, int out_size, void* d_ws, size_t ws_size,
                              hipStream_t stream) {
    const float* x      = (const float*)d_in[0];   // (32, 2048)
    const float* cached = (const float*)d_in[1];   // (32, 4096, 2048)
    const float* t      = (const float*)d_in[2];   // (32,)
    const float* wq     = (const float*)d_in[3];   // (128,)
    const float* bq     = (const float*)d_in[4];   // (128,)
    const float* wk     = (const float*)d_in[5];   // (128,)
    const float* bk     = (const float*)d_in[6];   // (128,)

    float* out     = (float*)d_out;                // [0 .. 65535]  : output (32,2048)
    float* weights = out + B_DIM * D_DIM;          // [65536 .. end]: weights (32,4096)
    float* qbuf    = (float*)d_ws;                 // 32*256 f32 = 32KB scratch

    // A: build query (cos|sin) halves
    ea_query_kernel<<<B_DIM, DH, 0, stream>>>(x, t, wq, bq, qbuf);
    // B: raw scores straight into the weights output region
    ea_scores_kernel<<<B_DIM * 64, 256, 0, stream>>>(cached, wk, bk, qbuf, weights);
    // C: softmax in place
    ea_softmax_kernel<<<B_DIM, 256, 0, stream>>>(weights);
    // D: weighted sum over full 2048-dim states via f32 WMMA
    ea_output_wmma_kernel<<<B_DIM * 16, 256, 0, stream>>>(cached, weights, out);
}